// GraphAngleProcessor_64012192579964
// MI455X (gfx1250) — compile-verified
//
#include <hip/hip_runtime.h>
#include <math.h>

#define THREADS 256
#define PAIRS_PER_THREAD 8
#define TILE (THREADS * PAIRS_PER_THREAD)   // 2048 pairs per block

#ifndef __has_builtin
#define __has_builtin(x) 0
#endif

#if __has_builtin(__builtin_amdgcn_global_load_async_to_lds_b128) && \
    __has_builtin(__builtin_amdgcn_s_wait_asynccnt)
#define ASYNC_LDS 1
typedef int v4i __attribute__((vector_size(16)));
typedef __attribute__((address_space(1))) v4i GV4;   // global
typedef __attribute__((address_space(3))) v4i LV4;   // LDS
// CPol: low 3 bits = temporal hint; 1 = NT (evict-first, protect L2 residency
// of the hot gather table from the streaming index traffic).
#define CPOL_NT 1
#else
#define ASYNC_LDS 0
#endif

typedef float v4f __attribute__((ext_vector_type(4)));

// --- Kernel 1: dir[e] = vec[e] / max(dist[e], 1e-5), padded to float4 ------
// dir table (32MB) is written with default RT hint: we WANT it resident in
// the 192MB L2 for the 40M random gathers that follow.
__global__ void ga_normalize_kernel(const float* __restrict__ dist,
                                    const float* __restrict__ vec,
                                    float4* __restrict__ dir, int E) {
    int e = blockIdx.x * blockDim.x + threadIdx.x;
    if (e >= E) return;
    float d = dist[e];
    float r = 1.0f / fmaxf(d, 1e-5f);
    float x = vec[3 * e + 0];
    float y = vec[3 * e + 1];
    float z = vec[3 * e + 2];
    dir[e] = make_float4(x * r, y * r, z * r, 0.0f);
}

// --- Kernel 2: full tiles, async-to-LDS index staging ----------------------
__global__ void ga_angles_tile_kernel(const int* __restrict__ asrc,
                                      const int* __restrict__ adst,
                                      const float4* __restrict__ dir,
                                      float* __restrict__ out) {
    __shared__ alignas(16) int s_src[TILE];
    __shared__ alignas(16) int s_dst[TILE];

    const long long base = (long long)blockIdx.x * TILE;
    const int t = threadIdx.x;

#if ASYNC_LDS
    // 2048 ints per array = 512 b128 transfers; 256 threads issue 2 each.
    // NT hint: indices are touched exactly once -> don't displace dir table.
#pragma unroll
    for (int k = 0; k < 2; ++k) {
        int idx4 = (t + k * THREADS) * 4;
        __builtin_amdgcn_global_load_async_to_lds_b128(
            (GV4*)(asrc + base + idx4), (LV4*)(s_src + idx4), 0, CPOL_NT);
        __builtin_amdgcn_global_load_async_to_lds_b128(
            (GV4*)(adst + base + idx4), (LV4*)(s_dst + idx4), 0, CPOL_NT);
    }
    // Prefetch next tile's indices into L2 while the async copies land.
    // Speculative prefetch past the end of the array is silently dropped.
    __builtin_prefetch(asrc + base + TILE + (long long)t * 4, 0, 3);
    __builtin_prefetch(adst + base + TILE + (long long)t * 4, 0, 3);
    __builtin_amdgcn_s_wait_asynccnt(0);
    __syncthreads();
#else
#pragma unroll
    for (int k = 0; k < 2; ++k) {
        int idx4 = (t + k * THREADS) * 4;
        *(int4*)(s_src + idx4) = *(const int4*)(asrc + base + idx4);
        *(int4*)(s_dst + idx4) = *(const int4*)(adst + base + idx4);
    }
    __syncthreads();
#endif

    float res[PAIRS_PER_THREAD];
#pragma unroll
    for (int j = 0; j < PAIRS_PER_THREAD; ++j) {
        int s = s_src[t * PAIRS_PER_THREAD + j];
        int d = s_dst[t * PAIRS_PER_THREAD + j];
        float4 a = dir[s];     // global_load_b128, RT hint: L2-resident table
        float4 b = dir[d];
        float c = a.x * b.x + a.y * b.y + a.z * b.z;
        res[j] = acosf(0.95f * c);
    }
#pragma unroll
    for (int j = 0; j < PAIRS_PER_THREAD; j += 4) {
        v4f o = {res[j], res[j + 1], res[j + 2], res[j + 3]};
        __builtin_nontemporal_store(
            o, (v4f*)(out + base + (long long)t * PAIRS_PER_THREAD + j));
    }
}

// --- Kernel 3: tail (< TILE pairs) -----------------------------------------
__global__ void ga_angles_tail_kernel(const int* __restrict__ asrc,
                                      const int* __restrict__ adst,
                                      const float4* __restrict__ dir,
                                      float* __restrict__ out,
                                      long long start, long long n) {
    long long i = start + (long long)blockIdx.x * blockDim.x + threadIdx.x;
    if (i >= n) return;
    int s = asrc[i];
    int d = adst[i];
    float4 a = dir[s];
    float4 b = dir[d];
    out[i] = acosf(0.95f * (a.x * b.x + a.y * b.y + a.z * b.z));
}

// --- Fallback if workspace is too small for the 32MB dir table -------------
__global__ void ga_angles_fly_kernel(const float* __restrict__ dist,
                                     const float* __restrict__ vec,
                                     const int* __restrict__ asrc,
                                     const int* __restrict__ adst,
                                     float* __restrict__ out, long long n) {
    long long i = (long long)blockIdx.x * blockDim.x + threadIdx.x;
    if (i >= n) return;
    int s = asrc[i];
    int d = adst[i];
    float rs = 1.0f / fmaxf(dist[s], 1e-5f);
    float rd = 1.0f / fmaxf(dist[d], 1e-5f);
    float c = (vec[3 * s + 0] * vec[3 * d + 0] +
               vec[3 * s + 1] * vec[3 * d + 1] +
               vec[3 * s + 2] * vec[3 * d + 2]) * rs * rd;
    out[i] = acosf(0.95f * c);
}

extern "C" void kernel_launch(void* const* d_in, const int* in_sizes, int n_in,
                              void* d_out, int out_size, void* d_ws, size_t ws_size,
                              hipStream_t stream) {
    const float* dist = (const float*)d_in[0];   // [E]
    const float* vec  = (const float*)d_in[1];   // [E,3]
    const int*   asrc = (const int*)d_in[2];     // [A]
    const int*   adst = (const int*)d_in[3];     // [A]
    float* out = (float*)d_out;                  // [A]

    const long long E = in_sizes[0];
    const long long A = in_sizes[2];

    if (ws_size >= (size_t)E * sizeof(float4)) {
        float4* dir = (float4*)d_ws;
        {
            int blocks = (int)((E + THREADS - 1) / THREADS);
            ga_normalize_kernel<<<blocks, THREADS, 0, stream>>>(dist, vec, dir, (int)E);
        }
        long long nTiles = A / TILE;
        if (nTiles > 0) {
            ga_angles_tile_kernel<<<(unsigned)nTiles, THREADS, 0, stream>>>(
                asrc, adst, dir, out);
        }
        long long done = nTiles * TILE;
        long long rem = A - done;
        if (rem > 0) {
            int blocks = (int)((rem + THREADS - 1) / THREADS);
            ga_angles_tail_kernel<<<blocks, THREADS, 0, stream>>>(
                asrc, adst, dir, out, done, A);
        }
    } else {
        int blocks = (int)((A + THREADS - 1) / THREADS);
        ga_angles_fly_kernel<<<blocks, THREADS, 0, stream>>>(
            dist, vec, asrc, adst, out, A);
    }
}